// FPSSampler_9732395893060
// MI455X (gfx1250) — compile-verified
//
#include <hip/hip_runtime.h>
#include <hip/hip_bf16.h>
#include <stdint.h>

// ---------------------------------------------------------------------------
// Furthest Point Sampling on MI455X (gfx1250)
//   B=4 batches, N=131072 points (xyz f32), K=2048 keypoints.
//   16 workgroups per batch; each WG keeps its 8192-point tile (96KB) and the
//   running min-dist array (32KB) resident in LDS (320KB/WGP on CDNA5).
//   Staging uses the Tensor Data Mover (tensor_load_to_lds + s_wait_tensorcnt,
//   clang-23 6-arg builtin form). Per-iteration cross-WG sync uses a monotonic
//   agent-scope atomic counter with release/acquire ordering; partial argmax
//   results are exchanged through a double-buffered slot array in workspace.
// ---------------------------------------------------------------------------

#define FPS_B    4
#define FPS_N    131072
#define FPS_K    2048
#define FPS_WGS  16                      // workgroups per batch
#define FPS_PPW  (FPS_N / FPS_WGS)       // 8192 points per WG
#define FPS_T    256                     // threads per WG (8 waves, wave32)
#define FPS_PPT  (FPS_PPW / FPS_T)       // 32 points per thread
#define FPS_NWAVE (FPS_T / 32)

typedef unsigned int u32x4 __attribute__((ext_vector_type(4)));
typedef int          i32x8 __attribute__((ext_vector_type(8)));
typedef int          i32x4 __attribute__((ext_vector_type(4)));

// Zero the per-batch barrier counters every launch (graph-replay safe:
// no reliance on prior workspace state).
__global__ void fps_init_kernel(int* __restrict__ counters) {
    if (threadIdx.x < FPS_B) counters[threadIdx.x] = 0;
}

__launch_bounds__(FPS_T, 1)
__global__ void fps_main_kernel(const float* __restrict__ pts,
                                float* __restrict__ out,
                                float* __restrict__ partials,
                                int* __restrict__ counters)
{
    __shared__ float s_pts[FPS_PPW * 3];   // 96 KB: this WG's point tile
    __shared__ float s_min[FPS_PPW];       // 32 KB: running min squared dist
    __shared__ float s_rv[FPS_NWAVE];
    __shared__ int   s_ri[FPS_NWAVE];
    __shared__ float s_q[3];               // broadcast of current query point

    const int b      = blockIdx.x / FPS_WGS;
    const int wg     = blockIdx.x % FPS_WGS;
    const int tid    = threadIdx.x;
    const int wgBase = wg * FPS_PPW;                       // local idx base in batch
    const float* batchPts = pts + (size_t)b * FPS_N * 3;
    const float* tile     = batchPts + (size_t)wgBase * 3; // 98304 contiguous bytes

    // Init running min-dists while the TDM streams points in.
    for (int i = tid; i < FPS_PPW; i += FPS_T) s_min[i] = 1e10f;

#if __has_builtin(__builtin_amdgcn_tensor_load_to_lds) && __has_builtin(__builtin_amdgcn_s_wait_tensorcnt)
    {
        // Each of the 8 waves DMAs its own 1024-point (12288 B = 3072 dword)
        // chunk: 1D-degenerate 2D tile, data_size = 4 bytes.
        const int      wave  = tid >> 5;                   // uniform per wave
        const unsigned bytesPerWave = (FPS_PPW / FPS_NWAVE) * 12u;   // 12288
        const unsigned DW = bytesPerWave / 4u;                        // 3072
        uint64_t gaddr = (uint64_t)(uintptr_t)tile + (uint64_t)wave * bytesPerWave;
        uint32_t laddr = (uint32_t)(uintptr_t)(&s_pts[0]) + (uint32_t)wave * bytesPerWave;
        // Force descriptor fields into SGPRs (wave-uniform by construction).
        uint32_t ga_lo = (uint32_t)__builtin_amdgcn_readfirstlane((int)(gaddr & 0xFFFFFFFFu));
        uint32_t ga_hi = (uint32_t)__builtin_amdgcn_readfirstlane((int)(gaddr >> 32));
        uint32_t la    = (uint32_t)__builtin_amdgcn_readfirstlane((int)laddr);

        u32x4 g0;
        g0[0] = 1u;                                  // count=1 (valid user D#)
        g0[1] = la;                                  // lds_addr
        g0[2] = ga_lo;                               // global_addr[31:0]
        g0[3] = (ga_hi & 0x01FFFFFFu) | (2u << 30);  // global_addr[56:32] | type=2
        i32x8 g1;
        g1[0] = (int)(2u << 16);                     // wg_mask=0, data_size=2 (4B)
        g1[1] = (int)(DW << 16);                     // tensor_dim0[15:0]=3072
        g1[2] = (int)(1u << 16);                     // tensor_dim0 hi=0 | tensor_dim1=1
        g1[3] = (int)(DW << 16);                     // tensor_dim1 hi=0 | tile_dim0=3072
        g1[4] = 1;                                   // tile_dim1=1, tile_dim2=0
        g1[5] = (int)DW;                             // tensor_dim0_stride lo
        g1[6] = (int)(DW << 16);                     // stride0 hi=0 | tensor_dim1_stride lo
        g1[7] = 0;                                   // tensor_dim1_stride hi
        i32x4 g2 = (i32x4){0, 0, 0, 0};
        i32x4 g3 = (i32x4){0, 0, 0, 0};
        i32x8 g4 = (i32x8){0, 0, 0, 0, 0, 0, 0, 0}; // extra group (clang-23 form)
        __builtin_amdgcn_tensor_load_to_lds(g0, g1, g2, g3, g4, 0);
        __builtin_amdgcn_s_wait_tensorcnt(0);        // TENSORcnt == 0 for this wave
    }
#else
    for (int i = tid; i < FPS_PPW * 3; i += FPS_T) s_pts[i] = tile[i];
#endif
    __syncthreads();

    // First selected point is index 0 of the batch (reference starts last_idx=0).
    float qx = batchPts[0], qy = batchPts[1], qz = batchPts[2];
    if (wg == 0 && tid == 0) {
        size_t o = (size_t)b * FPS_K * 3;
        out[o + 0] = qx; out[o + 1] = qy; out[o + 2] = qz;
    }

    float* partBase = partials + (size_t)b * 2 * FPS_WGS * 8;  // [phase][wg][8]
    int*   cnt      = counters + b;

    for (int k = 1; k < FPS_K; ++k) {
        // --- local distance update + per-thread argmax over 32 LDS points ---
        float bv = -1.0f; int bi = 0x7FFFFFFF;
        #pragma unroll 4
        for (int t = 0; t < FPS_PPT; ++t) {
            int   j  = tid + t * FPS_T;                 // conflict-free LDS strides
            float dx = s_pts[3 * j + 0] - qx;
            float dy = s_pts[3 * j + 1] - qy;
            float dz = s_pts[3 * j + 2] - qz;
            float d  = dx * dx + dy * dy + dz * dz;
            float m  = fminf(s_min[j], d);
            s_min[j] = m;
            if (m > bv) { bv = m; bi = j; }             // strict > keeps lowest j
        }
        // --- wave32 shuffle argmax (lowest index wins ties, like jnp.argmax) ---
        #pragma unroll
        for (int off = 16; off > 0; off >>= 1) {
            float ov = __shfl_down(bv, (unsigned)off, 32);
            int   oi = __shfl_down(bi, (unsigned)off, 32);
            if (ov > bv || (ov == bv && oi < bi)) { bv = ov; bi = oi; }
        }
        if ((tid & 31) == 0) { s_rv[tid >> 5] = bv; s_ri[tid >> 5] = bi; }
        __syncthreads();

        if (tid == 0) {
            #pragma unroll
            for (int w = 1; w < FPS_NWAVE; ++w) {
                float ov = s_rv[w]; int oi = s_ri[w];
                if (ov > bv || (ov == bv && oi < bi)) { bv = ov; bi = oi; }
            }
            // Publish this WG's partial (double-buffered by iteration parity).
            float* slot = partBase + (size_t)(k & 1) * FPS_WGS * 8 + wg * 8;
            slot[0] = bv;
            ((int*)slot)[1] = wgBase + bi;
            slot[2] = s_pts[3 * bi + 0];
            slot[3] = s_pts[3 * bi + 1];
            slot[4] = s_pts[3 * bi + 2];
            // Release: order the slot stores before the arrival increment.
            __hip_atomic_fetch_add(cnt, 1, __ATOMIC_RELEASE, __HIP_MEMORY_SCOPE_AGENT);
            const int target = FPS_WGS * k;             // monotonic, never reset
            while (__hip_atomic_load(cnt, __ATOMIC_ACQUIRE, __HIP_MEMORY_SCOPE_AGENT) < target) {
                __builtin_amdgcn_s_sleep(1);
            }
            // All 16 partials visible (acquire). Pick the global winner; wg
            // order ascending + strict > reproduces first-max tie-breaking.
            const float* ph = partBase + (size_t)(k & 1) * FPS_WGS * 8;
            float wv = -1.0f; int wi = 0x7FFFFFFF; float wx = 0, wy = 0, wz = 0;
            #pragma unroll
            for (int g = 0; g < FPS_WGS; ++g) {
                const float* s = ph + g * 8;
                float v = s[0]; int i2 = ((const int*)s)[1];
                if (v > wv || (v == wv && i2 < wi)) {
                    wv = v; wi = i2; wx = s[2]; wy = s[3]; wz = s[4];
                }
            }
            s_q[0] = wx; s_q[1] = wy; s_q[2] = wz;
            if (wg == 0) {
                size_t o = ((size_t)b * FPS_K + k) * 3;
                out[o + 0] = wx; out[o + 1] = wy; out[o + 2] = wz;
            }
        }
        __syncthreads();
        qx = s_q[0]; qy = s_q[1]; qz = s_q[2];
    }
}

extern "C" void kernel_launch(void* const* d_in, const int* in_sizes, int n_in,
                              void* d_out, int out_size, void* d_ws, size_t ws_size,
                              hipStream_t stream) {
    const float* pts = (const float*)d_in[0];           // [B,N,3] f32
    // d_in[1] = num_keypoints (scalar 2048, fixed by the reference setup)
    float* out      = (float*)d_out;                    // [B,K,3] f32
    float* partials = (float*)d_ws;                     // B*2*16*8 floats = 4 KB
    int*   counters = (int*)((char*)d_ws + (size_t)FPS_B * 2 * FPS_WGS * 8 * sizeof(float));

    fps_init_kernel<<<1, 64, 0, stream>>>(counters);
    fps_main_kernel<<<FPS_B * FPS_WGS, FPS_T, 0, stream>>>(pts, out, partials, counters);
}